// EvolvedLoss_9105330667723
// MI455X (gfx1250) — compile-verified
//
#include <hip/hip_runtime.h>

#define ROWS 2048
#define COLS 32000
#define TPB  320                       // 10 wave32s
#define F4_PER_ROW (COLS / 4)          // 8000
#define ITERS (F4_PER_ROW / TPB)       // 25, exact
#define NWAVES (TPB / 32)              // 10
#define NBUF 4                         // async pipeline: prefetch distance 3

typedef __attribute__((ext_vector_type(2))) float v2f;
typedef __attribute__((ext_vector_type(8))) float v8f;
typedef int v4i_b __attribute__((vector_size(16)));   // matches builtin param

// ---------------------------------------------------------------------------
// Wave32 full reduction through the matrix engine:
// A (16x4 f32): VGPR0 = per-lane partial, VGPR1 = 0
//   -> A[m][0] = p[m] (lanes 0..15), A[m][2] = p[m+16] (lanes 16..31)
// B (4x16 f32) = all ones  =>  D[m][n] = p[m] + p[m+16]  (replicated over n)
// Sum D's 8 VGPRs per lane; one shfl_xor(16) finishes the 32-lane total.
// ---------------------------------------------------------------------------
__device__ __forceinline__ float waveReduceWMMA(float x) {
#if __has_builtin(__builtin_amdgcn_wmma_f32_16x16x4_f32)
  v2f a; a[0] = x;   a[1] = 0.0f;
  v2f b; b[0] = 1.0f; b[1] = 1.0f;
  v8f c = {};
  c = __builtin_amdgcn_wmma_f32_16x16x4_f32(
      /*neg_a=*/false, a, /*neg_b=*/false, b,
      /*c_mod=*/(short)0, c, /*reuse_a=*/false, /*reuse_b=*/false);
  float t = ((c[0] + c[1]) + (c[2] + c[3])) + ((c[4] + c[5]) + (c[6] + c[7]));
  t += __shfl_xor(t, 16, 32);
  return t;
#else
  for (int o = 16; o > 0; o >>= 1) x += __shfl_xor(x, o, 32);
  return x;
#endif
}

// ---------------------------------------------------------------------------
// Async global->LDS staging (CDNA5 path, ASYNCcnt). Each lane copies its own
// 16B and later reads back only its own slot, so no cross-thread barrier is
// needed; s_wait_asynccnt alone orders LDS visibility for the issuing wave.
// ---------------------------------------------------------------------------
#if __has_builtin(__builtin_amdgcn_global_load_async_to_lds_b128) && \
    __has_builtin(__builtin_amdgcn_s_wait_asynccnt)
#define GAS __attribute__((address_space(1)))
#define LAS __attribute__((address_space(3)))
#define ASYNC_CP16(ldsp, glp)                                              \
  __builtin_amdgcn_global_load_async_to_lds_b128(                          \
      (GAS v4i_b*)(glp), (LAS v4i_b*)(ldsp), 0, 0)
#define ASYNC_WAIT(n) __builtin_amdgcn_s_wait_asynccnt(n)
#else
#define ASYNC_CP16(ldsp, glp) (*(float4*)(ldsp) = *(const float4*)(glp))
#define ASYNC_WAIT(n)
#endif

struct Stage {
  float4 o[TPB];
  float4 t[TPB];
};

__global__ __launch_bounds__(TPB) void rowKernel(
    const float* __restrict__ outp, const float* __restrict__ targ,
    const float* __restrict__ c1, const float* __restrict__ c2,
    float* __restrict__ partial) {
  __shared__ Stage stage[NBUF];          // 4 * 320 * 32B = 40 KB
  __shared__ float sw[NWAVES];
  __shared__ float sMu;

  const int row  = blockIdx.x;
  const int tid  = threadIdx.x;
  const int wave = tid >> 5;
  const int lane = tid & 31;

  const float4* gO = (const float4*)outp + (size_t)row * F4_PER_ROW;
  const float4* gT = (const float4*)targ + (size_t)row * F4_PER_ROW;

  // ---------------- Pass A: rowsum of (o - t)^2, depth-3 async pipeline ---
#pragma unroll
  for (int p = 0; p < NBUF - 1; ++p) {   // preload stages 0..2
    ASYNC_CP16(&stage[p].o[tid], gO + p * TPB + tid);
    ASYNC_CP16(&stage[p].t[tid], gT + p * TPB + tid);
  }
  float s2 = 0.0f;
  for (int it = 0; it < ITERS; ++it) {
    const int buf = it & (NBUF - 1);
    if (it + 3 < ITERS) {                // steady state: 3 stages in flight
      const int ni = (it + 3) * TPB + tid;
      const int nb = (it + 3) & (NBUF - 1);
      ASYNC_CP16(&stage[nb].o[tid], gO + ni);
      ASYNC_CP16(&stage[nb].t[tid], gT + ni);
      ASYNC_WAIT(6);
    } else if (it + 3 == ITERS) {
      ASYNC_WAIT(4);
    } else if (it + 2 == ITERS) {
      ASYNC_WAIT(2);
    } else {
      ASYNC_WAIT(0);
    }
    const float4 ov = stage[buf].o[tid];
    const float4 tv = stage[buf].t[tid];
    const float d0 = ov.x - tv.x, d1 = ov.y - tv.y;
    const float d2 = ov.z - tv.z, d3 = ov.w - tv.w;
    s2 += d0 * d0 + d1 * d1 + d2 * d2 + d3 * d3;
  }

  const float wsum = waveReduceWMMA(s2);
  if (lane == 0) sw[wave] = wsum;
  __syncthreads();
  if (tid == 0) {
    float tot = 0.0f;
    for (int w = 0; w < NWAVES; ++w) tot += sw[w];
    sMu = tot / (float)COLS;
  }
  __syncthreads();

  const float mu = sMu;
  const float k  = c1[2] * c1[4];  // centered*c1[4] folds into (d^2-mu)*k
  const float bb = c2[4];          // c2[2] cancels in the row-centering

  // ---------------- Pass B: re-stream row (L2-resident), finish chain -----
  float sm = 0.0f;
#pragma unroll
  for (int p = 0; p < NBUF - 1; ++p) {
    ASYNC_CP16(&stage[p].o[tid], gO + p * TPB + tid);
    ASYNC_CP16(&stage[p].t[tid], gT + p * TPB + tid);
  }
  for (int it = 0; it < ITERS; ++it) {
    const int buf = it & (NBUF - 1);
    if (it + 3 < ITERS) {
      const int ni = (it + 3) * TPB + tid;
      const int nb = (it + 3) & (NBUF - 1);
      ASYNC_CP16(&stage[nb].o[tid], gO + ni);
      ASYNC_CP16(&stage[nb].t[tid], gT + ni);
      ASYNC_WAIT(6);
    } else if (it + 3 == ITERS) {
      ASYNC_WAIT(4);
    } else if (it + 2 == ITERS) {
      ASYNC_WAIT(2);
    } else {
      ASYNC_WAIT(0);
    }
    const float4 ov = stage[buf].o[tid];
    const float4 tv = stage[buf].t[tid];
    const float d[4] = {ov.x - tv.x, ov.y - tv.y, ov.z - tv.z, ov.w - tv.w};
#pragma unroll
    for (int j = 0; j < 4; ++j) {
      const float arg = fmaf(fmaf(d[j], d[j], -mu), k, bb);
      sm += log1pf(fabsf(tanhf(arg)));
    }
  }

  const float wsumB = waveReduceWMMA(sm);
  if (lane == 0) sw[wave] = wsumB;
  __syncthreads();
  if (tid == 0) {
    float tot = 0.0f;
    for (int w = 0; w < NWAVES; ++w) tot += sw[w];
    partial[row] = tot;
  }
}

__global__ __launch_bounds__(256) void finalKernel(
    const float* __restrict__ partial, float* __restrict__ out) {
  __shared__ float sw[8];
  const int tid = threadIdx.x;
  float s = 0.0f;
  for (int i = tid; i < ROWS; i += 256) s += partial[i];
  const float w = waveReduceWMMA(s);
  const int wave = tid >> 5, lane = tid & 31;
  if (lane == 0) sw[wave] = w;
  __syncthreads();
  if (tid == 0) {
    float tot = 0.0f;
    for (int i = 0; i < 8; ++i) tot += sw[i];
    out[0] = tot / ((float)ROWS * (float)COLS);
  }
}

extern "C" void kernel_launch(void* const* d_in, const int* in_sizes, int n_in,
                              void* d_out, int out_size, void* d_ws,
                              size_t ws_size, hipStream_t stream) {
  const float* outputs = (const float*)d_in[0];
  const float* targets = (const float*)d_in[1];
  const float* c1      = (const float*)d_in[2];
  const float* c2      = (const float*)d_in[3];
  float* partial = (float*)d_ws;  // ROWS floats, fully rewritten every call

  rowKernel<<<ROWS, TPB, 0, stream>>>(outputs, targets, c1, c2, partial);
  finalKernel<<<1, 256, 0, stream>>>(partial, (float*)d_out);
}